// ElasticEmbedding_53171695125093
// MI455X (gfx1250) — compile-verified
//
#include <hip/hip_runtime.h>
#include <hip/hip_bf16.h>

// ElasticEmbedding gather for MI455X (gfx1250, wave32).
//
// Bandwidth-bound: 134MB out + row reads (tables are 51MB+10MB -> L2-resident
// in 192MB L2). One wave32 per token: 128 floats = one float4 per lane (512B
// fully coalesced global_load_b128 / NT global_store_b128).
//
// Membership test: build remap[v] (-1 or residual row) once per call in d_ws
// (400KB, L2-hot) -> one dependent scalar load per token instead of a ~15-deep
// dependent binary-search chain. Binary-search fallback if ws_size too small.

typedef float __attribute__((ext_vector_type(4))) v4f;  // native vector: OK for NT builtins

#define TOKENS_PER_ITER 2

// ---- Kernel 1: remap[i] = -1 for all vocab ids -----------------------------
__global__ __launch_bounds__(256) void ElasticEmbedding_init_remap(
    int* __restrict__ remap, int V)
{
    int i = blockIdx.x * blockDim.x + threadIdx.x;
    if (i < V) remap[i] = -1;
}

// ---- Kernel 2: remap[ridx[i]] = i ------------------------------------------
__global__ __launch_bounds__(256) void ElasticEmbedding_scatter_remap(
    const int* __restrict__ ridx, int* __restrict__ remap, int R)
{
    int i = blockIdx.x * blockDim.x + threadIdx.x;
    if (i < R) remap[ridx[i]] = i;
}

// ---- Kernel 3: LUT gather (main path) --------------------------------------
__global__ __launch_bounds__(256) void ElasticEmbedding_53171695125093_kernel(
    const int* __restrict__ x,            // [n_tokens] vocab ids
    const int* __restrict__ remap,        // [V] residual row or -1
    const float* __restrict__ pre,        // [V, 128]
    const float* __restrict__ res,        // [R, 128]
    float* __restrict__ out,              // [n_tokens, 128]
    int n_tokens)
{
    const int lane            = threadIdx.x & 31;
    const int wave_in_block   = threadIdx.x >> 5;
    const int waves_per_block = blockDim.x >> 5;
    const int wave_id         = blockIdx.x * waves_per_block + wave_in_block;
    const int n_waves         = gridDim.x * waves_per_block;
    const int stride          = n_waves * TOKENS_PER_ITER;

    for (int t0 = wave_id * TOKENS_PER_ITER; t0 < n_tokens; t0 += stride) {
        if (t0 + stride < n_tokens) {
            __builtin_prefetch(&x[t0 + stride], 0, 1);   // global_prefetch_b8
        }

#pragma unroll
        for (int k = 0; k < TOKENS_PER_ITER; ++k) {
            const int t = t0 + k;
            if (t >= n_tokens) break;

            // Wave-uniform index chain kept in SGPRs: x[t] -> remap[v].
            const int v = __builtin_amdgcn_readfirstlane(x[t]);
            const int r = __builtin_amdgcn_readfirstlane(remap[v]);

            const v4f* __restrict__ srcp =
                (r >= 0) ? (const v4f*)(res + (size_t)r * 128)
                         : (const v4f*)(pre + (size_t)v * 128);

            const v4f val = srcp[lane];                  // global_load_b128 (RT, L2)

            v4f* __restrict__ dstp = (v4f*)(out + (size_t)t * 128);
            __builtin_nontemporal_store(val, &dstp[lane]);  // NT global_store_b128
        }
    }
}

// ---- Fallback: binary search over sorted residual_index --------------------
__global__ __launch_bounds__(256) void ElasticEmbedding_bsearch_kernel(
    const int* __restrict__ x,
    const int* __restrict__ ridx,
    const float* __restrict__ pre,
    const float* __restrict__ res,
    float* __restrict__ out,
    int n_tokens, int n_res)
{
    const int lane            = threadIdx.x & 31;
    const int wave_in_block   = threadIdx.x >> 5;
    const int waves_per_block = blockDim.x >> 5;
    const int wave_id         = blockIdx.x * waves_per_block + wave_in_block;
    const int n_waves         = gridDim.x * waves_per_block;

    for (int t = wave_id; t < n_tokens; t += n_waves) {
        const int v = __builtin_amdgcn_readfirstlane(x[t]);

        int lo = 0, hi = n_res - 1, r = -1;
        while (lo <= hi) {
            const int mid = (lo + hi) >> 1;
            const int rv  = ridx[mid];
            if (rv == v)      { r = mid; break; }
            else if (rv < v)  { lo = mid + 1; }
            else              { hi = mid - 1; }
        }

        const v4f* __restrict__ srcp =
            (r >= 0) ? (const v4f*)(res + (size_t)r * 128)
                     : (const v4f*)(pre + (size_t)v * 128);

        const v4f val = srcp[lane];
        v4f* __restrict__ dstp = (v4f*)(out + (size_t)t * 128);
        __builtin_nontemporal_store(val, &dstp[lane]);
    }
}

extern "C" void kernel_launch(void* const* d_in, const int* in_sizes, int n_in,
                              void* d_out, int out_size, void* d_ws, size_t ws_size,
                              hipStream_t stream) {
    const int*   x    = (const int*)d_in[0];      // x              [B*S]
    const int*   ridx = (const int*)d_in[1];      // residual_index [R]
    const float* pre  = (const float*)d_in[2];    // pretrained_embedding [V*D]
    const float* res  = (const float*)d_in[3];    // residual_embedding   [R*D]
    float*       out  = (float*)d_out;

    const int n_tokens = in_sizes[0];             // 262144
    const int n_res    = in_sizes[1];             // 20000
    const int D        = 128;
    const int V        = in_sizes[2] / D;         // 100000

    const int block  = 256;                       // 8 waves per block
    const int blocks = 2048;                      // 16384 waves

    if (ws_size >= (size_t)V * sizeof(int)) {
        int* remap = (int*)d_ws;

        ElasticEmbedding_init_remap<<<(V + block - 1) / block, block, 0, stream>>>(
            remap, V);
        ElasticEmbedding_scatter_remap<<<(n_res + block - 1) / block, block, 0, stream>>>(
            ridx, remap, n_res);
        ElasticEmbedding_53171695125093_kernel<<<blocks, block, 0, stream>>>(
            x, remap, pre, res, out, n_tokens);
    } else {
        ElasticEmbedding_bsearch_kernel<<<blocks, block, 0, stream>>>(
            x, ridx, pre, res, out, n_tokens, n_res);
    }
}